// MeanConv_53523882443592
// MI455X (gfx1250) — compile-verified
//
#include <hip/hip_runtime.h>
#include <hip/hip_bf16.h>

typedef __attribute__((ext_vector_type(2))) float v2f;
typedef __attribute__((ext_vector_type(4))) float v4f;
typedef __attribute__((ext_vector_type(8))) float v8f;

#define EMBED 128
#define LAYER 128

// ---------- K1: zero the count/offset array ----------
__global__ void zero_counts_kernel(int* __restrict__ cnt, int n) {
    int i = blockIdx.x * blockDim.x + threadIdx.x;
    if (i < n) cnt[i] = 0;
}

// ---------- K2: per-user degree count ----------
__global__ void count_edges_kernel(const int* __restrict__ rows, int* __restrict__ cnt, int E) {
    int e = blockIdx.x * blockDim.x + threadIdx.x;
    if (e < E) atomicAdd(&cnt[rows[e]], 1);
}

// ---------- K3: single-block exclusive scan -> offsets (in place) + cursor copy ----------
__global__ void __launch_bounds__(1024)
scan_offsets_kernel(int* __restrict__ cnt_off, int* __restrict__ cursor, int U, int E) {
    __shared__ int part[1024];
    const int t   = threadIdx.x;
    const int per = (U + 1023) >> 10;
    const int s   = t * per;
    int       e   = s + per; if (e > U) e = U;

    int sum = 0;
    for (int i = s; i < e; ++i) sum += cnt_off[i];
    part[t] = sum;
    __syncthreads();

    if (t == 0) {
        int run = 0;
        for (int i = 0; i < 1024; ++i) { int v = part[i]; part[i] = run; run += v; }
        cnt_off[U] = E;
    }
    __syncthreads();

    int run = part[t];
    for (int i = s; i < e; ++i) {
        int c = cnt_off[i];
        cnt_off[i] = run;     // exclusive offset
        cursor[i]  = run;     // scatter cursor
        run += c;
    }
}

// ---------- K4: scatter edges into CSR (col, val) ----------
__global__ void scatter_csr_kernel(const int* __restrict__ rows, const int* __restrict__ cols,
                                   const float* __restrict__ vals, int* __restrict__ cursor,
                                   int* __restrict__ csr_col, float* __restrict__ csr_val, int E) {
    int e = blockIdx.x * blockDim.x + threadIdx.x;
    if (e < E) {
        int r = rows[e];
        int p = atomicAdd(&cursor[r], 1);
        csr_col[p] = cols[e];
        csr_val[p] = vals[e];
    }
}

// ---------- K5: P = item_emb @ W  (fp32 WMMA 16x16x4) ----------
// Block = 256 thr = 8 waves. Block owns one 16-row M tile; wave w owns N tile w.
// A frag (16x4 f32, 2 VGPR): lanes 0-15 -> K=k,k+1 ; lanes 16-31 -> K=k+2,k+3.
// B frag mirrored over N. C/D: VGPR j, lanes 0-15 = row j, lanes 16-31 = row j+8.
__global__ void __launch_bounds__(256)
gemm_pw_kernel(const float* __restrict__ A, const float* __restrict__ W,
               float* __restrict__ P, int M) {
    const int wave = threadIdx.x >> 5;
    const int lane = threadIdx.x & 31;
    const int lid  = lane & 15;
    const int half = lane >> 4;
    const int m0   = blockIdx.x * 16;
    const int n0   = wave * 16;

    int rowA = m0 + lid;
    if (rowA > M - 1) rowA = M - 1;               // safe clamp (stores handled below)

    const float* aptr = A + (size_t)rowA * EMBED + 2 * half;       // + k per step
    const float* bptr = W + (size_t)(2 * half) * LAYER + n0 + lid; // + k*LAYER per step

    v8f c0 = {}; v8f c1 = {};
    for (int kb = 0; kb < 32; kb += 2) {
        const int k0 = kb * 4;
        v2f a, b;
        a.x = aptr[k0];            a.y = aptr[k0 + 1];
        b.x = bptr[k0 * LAYER];    b.y = bptr[(k0 + 1) * LAYER];
        c0 = __builtin_amdgcn_wmma_f32_16x16x4_f32(false, a, false, b, (short)0, c0, false, false);

        const int k1 = k0 + 4;
        a.x = aptr[k1];            a.y = aptr[k1 + 1];
        b.x = bptr[k1 * LAYER];    b.y = bptr[(k1 + 1) * LAYER];
        c1 = __builtin_amdgcn_wmma_f32_16x16x4_f32(false, a, false, b, (short)0, c1, false, false);
    }
    v8f c = c0 + c1;

    // Wave-uniform fast path: full tile -> one unguarded store clause (8x global_store_b32).
    float* pout = P + (size_t)(m0 + 8 * half) * LAYER + n0 + lid;
    if (m0 + 16 <= M) {
#pragma unroll
        for (int j = 0; j < 8; ++j) pout[(size_t)j * LAYER] = c[j];
    } else {
#pragma unroll
        for (int j = 0; j < 8; ++j) {
            if (m0 + 8 * half + j < M) pout[(size_t)j * LAYER] = c[j];
        }
    }
}

// ---------- K6: per-user aggregation: out[u] = n_j[u] * sum_e val_e * P[col_e] ----------
// One wave32 per user; lane holds float4 -> 32*4 = 128 columns. Gathers are 512B/edge,
// fully coalesced, from L2-resident P. 2x unrolled with dual accumulators to keep two
// gathers in flight per wave. Single 512B store per user, no float atomics.
__global__ void __launch_bounds__(256)
aggregate_kernel(const float* __restrict__ P, const int* __restrict__ offs,
                 const int* __restrict__ csr_col, const float* __restrict__ csr_val,
                 const float* __restrict__ n_j, float* __restrict__ out, int U) {
    const int wid  = threadIdx.x >> 5;
    const int lane = threadIdx.x & 31;
    const int u    = blockIdx.x * 8 + wid;
    if (u >= U) return;

    const int start = offs[u];
    const int end   = offs[u + 1];

    v4f acc0 = {};
    v4f acc1 = {};
    int j = start;
    for (; j + 1 < end; j += 2) {
        const int   ca = csr_col[j];
        const int   cb = csr_col[j + 1];
        const float va = csr_val[j];
        const float vb = csr_val[j + 1];
        const v4f pa = *(const v4f*)(P + (size_t)ca * LAYER + lane * 4);
        const v4f pb = *(const v4f*)(P + (size_t)cb * LAYER + lane * 4);
        acc0 += va * pa;
        acc1 += vb * pb;
    }
    if (j < end) {
        const int   ca = csr_col[j];
        const float va = csr_val[j];
        acc0 += va * (*(const v4f*)(P + (size_t)ca * LAYER + lane * 4));
    }
    const float nj = n_j[u];
    *(v4f*)(out + (size_t)u * LAYER + lane * 4) = (acc0 + acc1) * nj;
}

extern "C" void kernel_launch(void* const* d_in, const int* in_sizes, int n_in,
                              void* d_out, int out_size, void* d_ws, size_t ws_size,
                              hipStream_t stream) {
    const float* item_emb = (const float*)d_in[0];
    const float* user_nj  = (const float*)d_in[1];
    const float* W        = (const float*)d_in[2];
    const float* vals     = (const float*)d_in[3];
    const int*   rows     = (const int*)d_in[4];
    const int*   cols     = (const int*)d_in[5];
    // d_in[6] (n_users) is a device scalar; derive sizes from host-visible in_sizes instead.
    const int U = in_sizes[1];            // user_n_j has U elements
    const int E = in_sizes[3];            // adj_vals has E elements
    const int M = in_sizes[0] / EMBED;    // number of items

    // Workspace layout (P first: must be 16B aligned for float4 gathers)
    char*  ws  = (char*)d_ws;
    size_t off = 0;
    float* P       = (float*)(ws + off); off += (size_t)M * LAYER * sizeof(float);
    off = (off + 255) & ~(size_t)255;
    int*   cnt_off = (int*)(ws + off);   off += (size_t)(U + 1) * sizeof(int);
    off = (off + 255) & ~(size_t)255;
    int*   cursor  = (int*)(ws + off);   off += (size_t)U * sizeof(int);
    off = (off + 255) & ~(size_t)255;
    int*   csr_col = (int*)(ws + off);   off += (size_t)E * sizeof(int);
    off = (off + 255) & ~(size_t)255;
    float* csr_val = (float*)(ws + off); off += (size_t)E * sizeof(float);
    (void)ws_size; (void)n_in; (void)out_size;

    zero_counts_kernel<<<(U + 1 + 255) / 256, 256, 0, stream>>>(cnt_off, U + 1);
    count_edges_kernel<<<(E + 255) / 256, 256, 0, stream>>>(rows, cnt_off, E);
    scan_offsets_kernel<<<1, 1024, 0, stream>>>(cnt_off, cursor, U, E);
    scatter_csr_kernel<<<(E + 255) / 256, 256, 0, stream>>>(rows, cols, vals, cursor,
                                                            csr_col, csr_val, E);
    gemm_pw_kernel<<<(M + 15) / 16, 256, 0, stream>>>(item_emb, W, P, M);
    aggregate_kernel<<<(U + 7) / 8, 256, 0, stream>>>(P, cnt_off, csr_col, csr_val,
                                                      user_nj, (float*)d_out, U);
}